// BasicRecSys_6605659701821
// MI455X (gfx1250) — compile-verified
//
#include <hip/hip_runtime.h>

typedef __attribute__((ext_vector_type(2))) float v2f;
typedef __attribute__((ext_vector_type(4))) float v4f;
typedef __attribute__((ext_vector_type(8))) float v8f;

#define EMB 64
#define ROWS_PER_WAVE 16
#define THREADS 256   // 8 waves per block (wave32)

// One wave handles 16 batch rows. Because the WMMA B-operand is all-ones and
// the result is a sum over every K slot of every step, data->K-slot mapping is
// arbitrary: lanes 0-15 stream floats [0..31] of their row, lanes 16-31 stream
// floats [32..63], in contiguous float4 (b128) chunks. Each chunk pair is
// pre-reduced in packed VALU (pk_mul + pk_fma) to a float2, then one
// V_WMMA_F32_16X16X4_F32 with B=ones folds the 16x4 tile into the accumulator.
// D[i][j] = dot_i for all j; column N=0: rows 0..7 in lane 0 VGPR0..7,
// rows 8..15 in lane 16 VGPR0..7 -> two contiguous v8f stores.
__global__ __launch_bounds__(THREADS)
void recsys_dot_wmma(const int* __restrict__ users,
                     const int* __restrict__ movies,
                     const float* __restrict__ user_table,
                     const float* __restrict__ movie_table,
                     float* __restrict__ out,
                     int batch)
{
    const int lane        = threadIdx.x & 31;
    const int wave_global = blockIdx.x * (THREADS / 32) + (threadIdx.x >> 5);
    const int base        = wave_global * ROWS_PER_WAVE;
    if (base >= batch) return;                 // wave-uniform: EXEC stays all-1s

    const int r    = lane & 15;                // tile row served by this lane
    const int half = (lane >> 4) * (EMB / 2);  // row half: 0 or 32

    const int uidx = users[base + r];
    const int midx = movies[base + r];

    const float* __restrict__ up = user_table  + (size_t)uidx * EMB + half;
    const float* __restrict__ mp = movie_table + (size_t)midx * EMB + half;

    v8f acc = {};
    const v2f ones = {1.0f, 1.0f};

    // 8 chunks of float4 per half-row; fully unrolled so all 16 b128 gathers
    // issue up front (deep MLP), WMMA chain drains behind them.
#pragma unroll
    for (int k = 0; k < EMB / 2; k += 4) {
        v4f u = __builtin_nontemporal_load((const v4f*)(up + k));
        v4f m = __builtin_nontemporal_load((const v4f*)(mp + k));
        v2f p = __builtin_shufflevector(u, u, 0, 1) *
                __builtin_shufflevector(m, m, 0, 1);          // v_pk_mul_f32
        p    += __builtin_shufflevector(u, u, 2, 3) *
                __builtin_shufflevector(m, m, 2, 3);          // v_pk_fma_f32
        // (neg_a, A, neg_b, B, c_mod, C, reuse_a, reuse_b)
        acc = __builtin_amdgcn_wmma_f32_16x16x4_f32(
                  false, p, false, ones, (short)0, acc, false, false);
    }

    // Extract column N=0 (== dot products). base%16==0 -> 64B-aligned stores.
    if (lane == 0) {
        *(v8f*)(out + base) = acc;             // rows 0..7
    } else if (lane == 16) {
        *(v8f*)(out + base + 8) = acc;         // rows 8..15
    }
}

extern "C" void kernel_launch(void* const* d_in, const int* in_sizes, int n_in,
                              void* d_out, int out_size, void* d_ws, size_t ws_size,
                              hipStream_t stream) {
    const int*   users       = (const int*)  d_in[0];
    const int*   movies      = (const int*)  d_in[1];
    const float* user_table  = (const float*)d_in[2];
    const float* movie_table = (const float*)d_in[3];
    float*       out         = (float*)d_out;

    const int batch           = in_sizes[0];                   // 16384
    const int waves_needed    = (batch + ROWS_PER_WAVE - 1) / ROWS_PER_WAVE;
    const int waves_per_block = THREADS / 32;
    const int blocks          = (waves_needed + waves_per_block - 1) / waves_per_block;

    recsys_dot_wmma<<<blocks, THREADS, 0, stream>>>(
        users, movies, user_table, movie_table, out, batch);
}